// AttentiveFPReadout_7112465842769
// MI455X (gfx1250) — compile-verified
//
#include <hip/hip_runtime.h>
#include <hip/hip_bf16.h>
#include <math.h>

// ---------------------------------------------------------------------------
// AttentiveFP readout, MI455X (gfx1250, wave32).
// V=262144 nodes, G=4096 graphs, F=256 feats, T=2 steps.
// Memory-bound part: 3 streaming passes over node_feats (768 MB @ 23.3 TB/s).
// Compute part: small [4096 x 256] GEMMs done with V_WMMA_F32_16X16X4_F32.
// ---------------------------------------------------------------------------

typedef __attribute__((ext_vector_type(2))) float v2f;
typedef __attribute__((ext_vector_type(8))) float v8f;

__device__ __forceinline__ float softplus_(float x) {
  // stable log(1+exp(x)); the reference's -LOG2 shift cancels inside softmax
  return x > 0.0f ? x + log1pf(expf(-x)) : log1pf(expf(x));
}

// ---------------------------------------------------------------------------
// K0: segment boundaries. segment_ids is sorted; start[g] = lower_bound(g).
// start[G] == V automatically.
// ---------------------------------------------------------------------------
__global__ void k_bounds(const int* __restrict__ seg, int* __restrict__ start,
                         int V, int G) {
  int g = blockIdx.x * blockDim.x + threadIdx.x;
  if (g > G) return;
  int lo = 0, hi = V;
  while (lo < hi) {
    int mid = (lo + hi) >> 1;
    if (seg[mid] < g) lo = mid + 1; else hi = mid;
  }
  start[g] = lo;
}

// ---------------------------------------------------------------------------
// K1 (streaming pass 1, fused): per-graph feature sum (initial g_feats),
// z_node[t][v] = nf[v] . w_node[t] for t=0..T-1 (T<=2 here), and per-graph
// max of z_node (for the stable segment softmax later).
// One block per graph; thread = feature column (coalesced).
// ---------------------------------------------------------------------------
__global__ __launch_bounds__(256) void k_init(
    const float* __restrict__ nf, const int* __restrict__ start,
    const float* __restrict__ logit_w, float* __restrict__ g_feats,
    float* __restrict__ z_node, float* __restrict__ zmax_node,
    int V, int F, int G, int T) {
  int g = blockIdx.x;
  int tid = threadIdx.x;
  int s = start[g], e = start[g + 1];
  __shared__ float2 red[256];

  float wn0 = 0.0f, wn1 = 0.0f, acc = 0.0f;
  if (tid < F) {
    wn0 = logit_w[F + tid];                    // t=0: logit_w[0*2F + F + f]
    if (T > 1) wn1 = logit_w[3 * F + tid];     // t=1: logit_w[1*2F + F + f]
  }
  float zm0 = -1e30f, zm1 = -1e30f;

  for (int v = s; v < e; ++v) {
    float x = 0.0f;
    if (tid < F) {
      if ((v & 7) == 0) {                      // prefetch 8 rows ahead
        int pv = v + 8; if (pv >= V) pv = V - 1;
        __builtin_prefetch(&nf[(size_t)pv * F + tid], 0, 0);
      }
      x = nf[(size_t)v * F + tid];
    }
    acc += x;
    red[tid] = make_float2(x * wn0, x * wn1);
    __syncthreads();
    for (int st = 128; st > 0; st >>= 1) {
      if (tid < st) {
        red[tid].x += red[tid + st].x;
        red[tid].y += red[tid + st].y;
      }
      __syncthreads();
    }
    if (tid == 0) {
      float z0 = red[0].x, z1 = red[0].y;
      z_node[v] = z0;
      if (T > 1) z_node[(size_t)V + v] = z1;
      zm0 = fmaxf(zm0, z0);
      zm1 = fmaxf(zm1, z1);
    }
    __syncthreads();
  }
  if (tid < F) g_feats[(size_t)g * F + tid] = acc;
  if (tid == 0) {
    zmax_node[g] = zm0;
    if (T > 1) zmax_node[G + g] = zm1;
  }
}

// ---------------------------------------------------------------------------
// K2: per-graph logit scalar  zg[g] = relu(g_feats[g]) . w_ctx[t] + b[t].
// One wave32 per graph.
// ---------------------------------------------------------------------------
__global__ __launch_bounds__(256) void k_zg(
    const float* __restrict__ g_feats, const float* __restrict__ logit_w,
    const float* __restrict__ logit_b, float* __restrict__ zg,
    int G, int F, int t) {
  int gtid = blockIdx.x * blockDim.x + threadIdx.x;
  int wave = gtid >> 5, lane = gtid & 31;
  if (wave >= G) return;
  const float* h = g_feats + (size_t)wave * F;
  const float* wc = logit_w + (size_t)t * 2 * F;
  float acc = 0.0f;
  for (int f = lane; f < F; f += 32) acc += fmaxf(h[f], 0.0f) * wc[f];
  for (int m = 16; m > 0; m >>= 1) acc += __shfl_xor(acc, m, 32);
  if (lane == 0) zg[wave] = acc + logit_b[t];
}

// ---------------------------------------------------------------------------
// K3: segment-softmax denominator over the 1 MB z_node array (cheap).
// One wave32 per graph.
// ---------------------------------------------------------------------------
__global__ __launch_bounds__(256) void k_denom(
    const int* __restrict__ start, const float* __restrict__ z_node,
    const float* __restrict__ zmax_node, const float* __restrict__ zg,
    float* __restrict__ denom, int G, int V, int t) {
  int gtid = blockIdx.x * blockDim.x + threadIdx.x;
  int wave = gtid >> 5, lane = gtid & 31;
  if (wave >= G) return;
  int s = start[wave], e = start[wave + 1];
  float zgv = zg[wave];
  float qmax = softplus_(zgv + zmax_node[(size_t)t * G + wave]);
  const float* zn = z_node + (size_t)t * V;
  float acc = 0.0f;
  for (int v = s + lane; v < e; v += 32)
    acc += expf(softplus_(zgv + zn[v]) - qmax);
  for (int m = 16; m > 0; m >>= 1) acc += __shfl_xor(acc, m, 32);
  if (lane == 0) denom[wave] = acc;
}

// ---------------------------------------------------------------------------
// K4 (streaming pass 2/3): attention-weighted segment sum
//   wf[g] = sum_v a_v * nf[v],  a_v = exp(q(zg+zn_v) - qmax)/denom.
// Block per graph, thread = feature column. Attention weights for a chunk of
// rows are computed once into LDS, then the chunk is streamed.
// ---------------------------------------------------------------------------
__global__ __launch_bounds__(256) void k_wsum(
    const float* __restrict__ nf, const int* __restrict__ start,
    const float* __restrict__ z_node, const float* __restrict__ zmax_node,
    const float* __restrict__ zg, const float* __restrict__ denom,
    float* __restrict__ wf, int V, int F, int G, int t) {
  int g = blockIdx.x, tid = threadIdx.x;
  int s = start[g], e = start[g + 1];
  float zgv = zg[g];
  float qmax = softplus_(zgv + zmax_node[(size_t)t * G + g]);
  float invd = (e > s) ? 1.0f / denom[g] : 0.0f;
  const float* zn = z_node + (size_t)t * V;
  __shared__ float sa[256];
  float acc = 0.0f;
  for (int c = s; c < e; c += 256) {
    int n = e - c; if (n > 256) n = 256;
    if (tid < n) sa[tid] = expf(softplus_(zgv + zn[c + tid]) - qmax) * invd;
    __syncthreads();
    for (int r = 0; r < n; ++r) {
      if (tid < F) {
        if ((r & 7) == 0) {
          int pv = c + r + 8; if (pv >= V) pv = V - 1;
          __builtin_prefetch(&nf[(size_t)pv * F + tid], 0, 0);
        }
        acc += sa[r] * nf[(size_t)(c + r) * F + tid];
      }
    }
    __syncthreads();
  }
  if (tid < F) wf[(size_t)g * F + tid] = acc;
}

// ---------------------------------------------------------------------------
// WMMA f32 GEMM: C[M,N] = A[M,K] @ B (+bias[n]) (+activation).
//   BT==0: B stored [K,N] row-major (proj_w:  C = A @ B)
//   BT==1: B stored [N,K] row-major (GRU:     C = A @ B^T)
//   ACT==1: ELU epilogue.
// One wave32 per 16x16 output tile, K consumed 4 at a time with
// V_WMMA_F32_16X16X4_F32. A-frag per ISA layout: lanes 0-15 hold K=kb+0/1,
// lanes 16-31 hold K=kb+2/3, row = lane&15; C/D: VGPR j -> M = j + 8*(lane>>4).
// ---------------------------------------------------------------------------
template <int BT, int ACT>
__global__ __launch_bounds__(256) void gemm_wmma(
    const float* __restrict__ A, const float* __restrict__ B,
    const float* __restrict__ bias, float* __restrict__ C,
    int M, int N, int K) {
  int gtid = blockIdx.x * blockDim.x + threadIdx.x;
  int wave = gtid >> 5;
  int lane = gtid & 31;
  int mt = M >> 4, nt = N >> 4;
  if (wave >= mt * nt) return;            // wave-uniform exit: EXEC stays full
  int tn = wave % nt, tm = wave / nt;
  int l15 = lane & 15;
  int hi = lane >> 4;                     // 0 or 1
  int kh = hi * 2;                        // K sub-offset for this half-wave
  int row = tm * 16 + l15;
  int col = tn * 16 + l15;

  float bv = bias ? bias[col] : 0.0f;
  v8f acc;
#pragma unroll
  for (int j = 0; j < 8; ++j) acc[j] = bv;

  const float* arow = A + (size_t)row * K;
  for (int kb = 0; kb < K; kb += 4) {
    v2f a = *(const v2f*)(arow + kb + kh);
    v2f b;
    if (BT) {
      b = *(const v2f*)(B + (size_t)col * K + kb + kh);
    } else {
      b.x = B[(size_t)(kb + kh) * N + col];
      b.y = B[(size_t)(kb + kh + 1) * N + col];
    }
    acc = __builtin_amdgcn_wmma_f32_16x16x4_f32(
        /*neg_a=*/false, a, /*neg_b=*/false, b,
        /*c_mod=*/(short)0, acc, /*reuse_a=*/false, /*reuse_b=*/false);
  }

  float* crow = C + (size_t)(tm * 16 + hi * 8) * N + col;
#pragma unroll
  for (int j = 0; j < 8; ++j) {
    float x = acc[j];
    if (ACT == 1) x = x > 0.0f ? x : expm1f(x);   // ELU
    crow[(size_t)j * N] = x;
  }
}

// ---------------------------------------------------------------------------
// K6: GRU gates (biases already folded into gi/gh by the GEMMs).
// ---------------------------------------------------------------------------
__global__ __launch_bounds__(256) void k_gate(
    const float* __restrict__ gi, const float* __restrict__ gh,
    const float* __restrict__ h, float* __restrict__ out, int G, int F) {
  int idx = blockIdx.x * blockDim.x + threadIdx.x;
  if (idx >= G * F) return;
  int g = idx / F, f = idx % F;
  size_t b = (size_t)g * 3 * F + f;
  float r = 1.0f / (1.0f + expf(-(gi[b] + gh[b])));
  float z = 1.0f / (1.0f + expf(-(gi[b + F] + gh[b + F])));
  float n = tanhf(gi[b + 2 * F] + r * gh[b + 2 * F]);
  out[idx] = (1.0f - z) * n + z * h[idx];
}

// ---------------------------------------------------------------------------
// Host launch
// ---------------------------------------------------------------------------
extern "C" void kernel_launch(void* const* d_in, const int* in_sizes, int n_in,
                              void* d_out, int out_size, void* d_ws,
                              size_t ws_size, hipStream_t stream) {
  const float* nf      = (const float*)d_in[0];
  const int*   seg     = (const int*)d_in[1];
  const float* logit_w = (const float*)d_in[3];
  const float* logit_b = (const float*)d_in[4];
  const float* proj_w  = (const float*)d_in[5];
  const float* proj_b  = (const float*)d_in[6];
  const float* w_ih    = (const float*)d_in[7];
  const float* w_hh    = (const float*)d_in[8];
  const float* b_ih    = (const float*)d_in[9];
  const float* b_hh    = (const float*)d_in[10];

  const int V = in_sizes[1];            // 262144
  const int T = in_sizes[4];            // logit_b is [T,1] -> 2
  const int F = in_sizes[6] / T;        // proj_b is [T,F] -> 256
  const int G = out_size / F;           // output [G,F]    -> 4096

  float* out = (float*)d_out;
  float* w = (float*)d_ws;
  float* g_feats = w;  w += (size_t)G * F;
  float* z_node  = w;  w += (size_t)T * V;
  float* zmaxn   = w;  w += (size_t)T * G;
  float* zg      = w;  w += G;
  float* denomv  = w;  w += G;
  float* wf      = w;  w += (size_t)G * F;
  float* ctx     = w;  w += (size_t)G * F;
  float* gi      = w;  w += (size_t)G * 3 * F;
  float* gh      = w;  w += (size_t)G * 3 * F;
  int*   start   = (int*)w;             // G+1 ints

  k_bounds<<<(G + 1 + 255) / 256, 256, 0, stream>>>(seg, start, V, G);
  k_init<<<G, 256, 0, stream>>>(nf, start, logit_w, g_feats, z_node, zmaxn,
                                V, F, G, T);

  for (int t = 0; t < T; ++t) {
    k_zg<<<(G * 32 + 255) / 256, 256, 0, stream>>>(g_feats, logit_w, logit_b,
                                                   zg, G, F, t);
    k_denom<<<(G * 32 + 255) / 256, 256, 0, stream>>>(start, z_node, zmaxn,
                                                      zg, denomv, G, V, t);
    k_wsum<<<G, 256, 0, stream>>>(nf, start, z_node, zmaxn, zg, denomv, wf,
                                  V, F, G, t);

    {  // context = elu(wf @ proj_w[t] + proj_b[t])   [G,F]
      int M = G, N = F, K = F;
      int waves = (M / 16) * (N / 16);
      gemm_wmma<0, 1><<<(waves * 32 + 255) / 256, 256, 0, stream>>>(
          wf, proj_w + (size_t)t * F * F, proj_b + (size_t)t * F, ctx, M, N, K);
    }
    {  // gi = ctx @ w_ih[t]^T + b_ih[t];  gh = h @ w_hh[t]^T + b_hh[t]  [G,3F]
      int M = G, N = 3 * F, K = F;
      int waves = (M / 16) * (N / 16);
      gemm_wmma<1, 0><<<(waves * 32 + 255) / 256, 256, 0, stream>>>(
          ctx, w_ih + (size_t)t * 3 * F * F, b_ih + (size_t)t * 3 * F, gi,
          M, N, K);
      gemm_wmma<1, 0><<<(waves * 32 + 255) / 256, 256, 0, stream>>>(
          g_feats, w_hh + (size_t)t * 3 * F * F, b_hh + (size_t)t * 3 * F, gh,
          M, N, K);
    }
    float* hout = (t == T - 1) ? out : g_feats;
    k_gate<<<((size_t)G * F + 255) / 256, 256, 0, stream>>>(gi, gh, g_feats,
                                                            hout, G, F);
  }
}